// Distance_kernel_69337952027158
// MI455X (gfx1250) — compile-verified
//
#include <hip/hip_runtime.h>
#include <math.h>

typedef __attribute__((ext_vector_type(2))) float v2f;
typedef __attribute__((ext_vector_type(8))) float v8f;

#define BATCH 128

// ---------------------------------------------------------------------------
// Stage 1: embedding  d[b, j] = cos/sin(x[b] * freq[j%32])
// ---------------------------------------------------------------------------
__global__ void embed_kernel(const float* __restrict__ x, float* __restrict__ d) {
    const int b = blockIdx.x;
    const int j = threadIdx.x;                 // 0..63
    const float wavelength = 0.638e-6f;
    const float ratio     = wavelength / 8e-6f * 0.5f;          // wl/pitch/2
    const float two_pi    = 6.28318530717958647692f;
    const float max_fre   = two_pi / wavelength;
    const float min_fre   = max_fre * sqrtf(1.0f - 2.0f * ratio * ratio);
    const int   f         = (j < 32) ? j : (j - 32);
    const float freq      = (max_fre - min_fre) * (1.0f / 32.0f) * (float)(f + 1) + min_fre;
    const float ds        = x[b] * freq;
    d[b * 64 + j] = (j < 32) ? cosf(ds) : sinf(ds);
}

// ---------------------------------------------------------------------------
// Stage 2: f32 WMMA GEMM.  out(M x N) = A(M x K) * W(K x N) + bias.
// One wave (32 lanes) per 16x16 output tile; V_WMMA_F32_16X16X4_F32.
// A 16x4 layout : lane 0-15 -> M=lane, v0=K+0, v1=K+1 ; lane 16-31 -> K+2,K+3
// B 4x16 layout : v_r holds row K=r (lanes 0-15) / K=r+2 (lanes 16-31), N=lane&15
// C/D 16x16     : v_r lanes 0-15 -> M=r, lanes 16-31 -> M=r+8, N=lane&15
// ---------------------------------------------------------------------------
__global__ void gemm_wmma_f32(const float* __restrict__ A,
                              const float* __restrict__ W,
                              const float* __restrict__ bias,
                              float* __restrict__ out,
                              int K, int N) {
    const int lane = threadIdx.x & 31;
    const int half = lane >> 4;            // 0 | 1
    const int l15  = lane & 15;
    const int row0 = blockIdx.x * 16;      // M tile
    const int col0 = blockIdx.y * 16;      // N tile

    v8f acc = {};
    const float* ap = A + (row0 + l15) * K + half * 2;
    const float* bp = W + (half * 2) * N + col0 + l15;

    #pragma unroll 4
    for (int k = 0; k < K; k += 4) {
        v2f a, b;
        a.x = ap[k];                       // A[m][k + half*2 + 0]
        a.y = ap[k + 1];                   // A[m][k + half*2 + 1]
        b.x = bp[(long)k * N];             // W[k + half*2 + 0][n]
        b.y = bp[(long)(k + 1) * N];       // W[k + half*2 + 1][n]
        acc = __builtin_amdgcn_wmma_f32_16x16x4_f32(
            /*neg_a=*/false, a, /*neg_b=*/false, b,
            /*c_mod=*/(short)0, acc, /*reuse_a=*/false, /*reuse_b=*/false);
    }

    const float bv = bias[col0 + l15];
    #pragma unroll
    for (int r = 0; r < 8; ++r) {
        out[(long)(row0 + r + half * 8) * N + col0 + l15] = acc[r] + bv;
    }
}

// ---------------------------------------------------------------------------
// Stage 3: per-row LayerNorm + tanh, in place.  One block per row.
// ---------------------------------------------------------------------------
__global__ void ln_tanh_kernel(float* __restrict__ x,
                               const float* __restrict__ g,
                               const float* __restrict__ beta,
                               int N) {
    const int row = blockIdx.x;
    const int tid = threadIdx.x;           // 256 threads
    float* xr = x + (long)row * N;

    float s = 0.f, s2 = 0.f;
    for (int i = tid; i < N; i += 256) {
        float v = xr[i];
        s += v; s2 += v * v;
    }
    __shared__ float r1[256], r2[256];
    r1[tid] = s; r2[tid] = s2;
    __syncthreads();
    for (int off = 128; off > 0; off >>= 1) {
        if (tid < off) { r1[tid] += r1[tid + off]; r2[tid] += r2[tid + off]; }
        __syncthreads();
    }
    const float mean = r1[0] / (float)N;
    const float var  = r2[0] / (float)N - mean * mean;
    const float rstd = rsqrtf(var + 1e-5f);
    for (int i = tid; i < N; i += 256) {
        float v = (xr[i] - mean) * rstd * g[i] + beta[i];
        xr[i] = tanhf(v);
    }
}

// ---------------------------------------------------------------------------
// Stage 4a: build the radial bin-index map for a given L (once per call).
// map[i*2L + j] = floor(dis(i,j) / (interval + 1e-4)) as u8 (255 = invalid).
// Reused across all B*C2 slices -> removes sqrt/floor from the hot path.
// ---------------------------------------------------------------------------
__global__ void build_idx_kernel(int L, unsigned char* __restrict__ map) {
    const int twoL = 2 * L;
    const int t = blockIdx.x * blockDim.x + threadIdx.x;
    if (t >= twoL * twoL) return;
    const int i = t / twoL;                // launch-time small, cheap enough here
    const int j = t - i * twoL;
    const float step = (float)twoL / (float)(twoL - 1);   // linspace(-L, L, 2L)
    const float axi  = -(float)L + (float)i * step;
    const float axj  = -(float)L + (float)j * step;
    const float dis  = sqrtf(axi * axi + axj * axj);
    // interval = max(dis)/L computed exactly as the reference's f32 grid max
    const float interval = sqrtf(2.0f * (float)L * (float)L) / (float)L;
    const int idx = (int)floorf(dis / (interval + 0.0001f));
    map[t] = (idx < L) ? (unsigned char)idx : (unsigned char)255;
}

// ---------------------------------------------------------------------------
// Stage 4b: circular expansion — the bandwidth-bound stage.
// out[b,c,i,j] = (map[i,j] < L) ? src[b*stride + colOff + c*L + map[i,j]] : 0
// All shape factors are powers of two -> pure shift/mask indexing.
// One float4 store per thread; exact grid (no loop).
// ---------------------------------------------------------------------------
__global__ void circular_expand_kernel(const float* __restrict__ src,
                                       int srcStride, int colOff,
                                       int lgC2, int lgL,
                                       const unsigned char* __restrict__ map,
                                       float* __restrict__ out) {
    const int lgTwoL = lgL + 1;
    const int lgJq   = lgTwoL - 2;                 // float4 groups per row
    const int L      = 1 << lgL;

    const unsigned t  = blockIdx.x * blockDim.x + threadIdx.x;
    const unsigned j4 = t & ((1u << lgJq) - 1u);
    unsigned r        = t >> lgJq;
    const unsigned i  = r & ((1u << lgTwoL) - 1u); r >>= lgTwoL;
    const unsigned c  = r & ((1u << lgC2) - 1u);
    const unsigned b  = r >> lgC2;

    const float* s = src + (size_t)b * srcStride + colOff + (c << lgL);

    // 4 consecutive u8 bin indices (map row is >=32B, j4*4 is 4B-aligned)
    const unsigned mi = *(const unsigned*)(map + (i << lgTwoL) + (j4 << 2));
    const int i0 = (int)(mi & 0xFF);
    const int i1 = (int)((mi >> 8) & 0xFF);
    const int i2 = (int)((mi >> 16) & 0xFF);
    const int i3 = (int)((mi >> 24) & 0xFF);

    float4 o;
    o.x = (i0 < L) ? s[i0] : 0.0f;
    o.y = (i1 < L) ? s[i1] : 0.0f;
    o.z = (i2 < L) ? s[i2] : 0.0f;
    o.w = (i3 < L) ? s[i3] : 0.0f;
    *(float4*)(out + (size_t)t * 4) = o;
}

// ---------------------------------------------------------------------------
// Host-side launch
// ---------------------------------------------------------------------------
extern "C" void kernel_launch(void* const* d_in, const int* in_sizes, int n_in,
                              void* d_out, int out_size, void* d_ws, size_t ws_size,
                              hipStream_t stream) {
    const float* x     = (const float*)d_in[0];
    const float* W1    = (const float*)d_in[1];
    const float* b1    = (const float*)d_in[2];
    const float* g1    = (const float*)d_in[3];
    const float* beta1 = (const float*)d_in[4];
    const float* W2    = (const float*)d_in[5];
    const float* b2    = (const float*)d_in[6];
    const float* g2    = (const float*)d_in[7];
    const float* beta2 = (const float*)d_in[8];
    const float* W3    = (const float*)d_in[9];
    const float* b3    = (const float*)d_in[10];
    const float* g3    = (const float*)d_in[11];
    const float* beta3 = (const float*)d_in[12];

    float* ws  = (float*)d_ws;
    float* d   = ws;                          // 128 x 64
    float* h   = d   + BATCH * 64;            // 128 x 512
    float* kv  = h   + BATCH * 512;           // 128 x 768
    float* kvb = kv  + BATCH * 768;           // 128 x 1536
    unsigned char* maps  = (unsigned char*)(kvb + BATCH * 1536);
    unsigned char* mapL64 = maps;             // L=64 : 128*128 = 16384 B
    unsigned char* mapL32 = maps + 16384;     // L=32 :  64*64  =  4096 B
    unsigned char* mapL16 = maps + 20480;     // L=16 :  32*32  =  1024 B

    // Stage 1: embedding (then the map builds, in order on the same stream)
    embed_kernel<<<BATCH, 64, 0, stream>>>(x, d);
    build_idx_kernel<<<(16384 + 255) / 256, 256, 0, stream>>>(64, mapL64);
    build_idx_kernel<<<(4096  + 255) / 256, 256, 0, stream>>>(32, mapL32);
    build_idx_kernel<<<(1024  + 255) / 256, 256, 0, stream>>>(16, mapL16);

    // Stage 2+3: GEMM -> LN -> tanh chain (all f32 WMMA)
    gemm_wmma_f32<<<dim3(BATCH / 16, 512 / 16),  32, 0, stream>>>(d,  W1, b1, h,   64,  512);
    ln_tanh_kernel<<<BATCH, 256, 0, stream>>>(h,   g1, beta1, 512);
    gemm_wmma_f32<<<dim3(BATCH / 16, 768 / 16),  32, 0, stream>>>(h,  W2, b2, kv,  512, 768);
    ln_tanh_kernel<<<BATCH, 256, 0, stream>>>(kv,  g2, beta2, 768);
    gemm_wmma_f32<<<dim3(BATCH / 16, 1536 / 16), 32, 0, stream>>>(kv, W3, b3, kvb, 768, 1536);
    ln_tanh_kernel<<<BATCH, 256, 0, stream>>>(kvb, g3, beta3, 1536);

    // Stage 4: six circular expansions. Output order: c1, c2, c3, c11, c22, c33
    float* out = (float*)d_out;
    struct Job { const float* src; int stride, coloff, lgC2, lgL; const unsigned char* map; long outoff; };
    const Job jobs[6] = {
        { kvb, 1536, 0,    3, 6, mapL64, 0L        },  // c1 : (B,8,128,128)
        { kvb, 1536, 512,  4, 5, mapL32, 16777216L },  // c2 : (B,16,64,64)
        { kvb, 1536, 1024, 5, 4, mapL16, 25165824L },  // c3 : (B,32,32,32)
        { kv,  768,  0,    2, 6, mapL64, 29360128L },  // c11: (B,4,128,128)
        { kv,  768,  256,  3, 5, mapL32, 37748736L },  // c22: (B,8,64,64)
        { kv,  768,  512,  4, 4, mapL16, 41943040L },  // c33: (B,16,32,32)
    };
    for (int q = 0; q < 6; ++q) {
        const Job& J = jobs[q];
        const int twoL = 2 << J.lgL;
        long total4 = (long)BATCH * (1 << J.lgC2) * twoL * (twoL / 4);
        int  blocks = (int)(total4 / 256);     // all counts are multiples of 256
        circular_expand_kernel<<<blocks, 256, 0, stream>>>(
            J.src, J.stride, J.coloff, J.lgC2, J.lgL, J.map, out + J.outoff);
    }
}